// CosineDistanceLoss_35708358099383
// MI455X (gfx1250) — compile-verified
//
#include <hip/hip_runtime.h>
#include <math.h>

typedef __attribute__((ext_vector_type(2))) float v2f;
typedef __attribute__((ext_vector_type(8))) float v8f;

#define EPT 32          // contiguous elements per thread (batch_map sorted => few seg switches)
#define TPB 256
#define NSEG 16384

// ---------------------------------------------------------------------------
// Stage 1: bandwidth-bound segmented reduction.
// Each thread owns a contiguous 32-element run, loads p/t as float4 and
// batch_map as int4 (b128 loads), accumulates pp/tt/pt locally while the
// segment id is unchanged, and flushes with f32 global atomics on change.
// ---------------------------------------------------------------------------
__global__ __launch_bounds__(TPB) void seg_reduce_kernel(
    const float* __restrict__ p,
    const float* __restrict__ t,
    const int*   __restrict__ bm,
    float* __restrict__ seg_pp,
    float* __restrict__ seg_tt,
    float* __restrict__ seg_pt,
    int n) {
  long base = (long)(blockIdx.x * TPB + threadIdx.x) * EPT;
  if (base >= n) return;

  // Prefetch well ahead of this thread's run (gfx1250 global_prefetch_b8).
  if (base + 8192 < n) {
    __builtin_prefetch(p + base + 8192, 0, 0);
    __builtin_prefetch(t + base + 8192, 0, 0);
  }

  int   cur = bm[base];
  float app = 0.0f, att = 0.0f, apt = 0.0f;
  int   count = (int)((base + EPT <= n) ? EPT : (n - base));

  if (count == EPT) {
    const float4* p4 = (const float4*)(p + base);
    const float4* t4 = (const float4*)(t + base);
    const int4*   m4 = (const int4*)(bm + base);
#pragma unroll
    for (int g = 0; g < EPT / 4; ++g) {
      float4 pv = p4[g];
      float4 tv = t4[g];
      int4   mv = m4[g];
      if (mv.x == cur && mv.w == cur) {
        // Fast path: whole quad in the current segment (sorted => monotonic).
        app = fmaf(pv.x, pv.x, app); att = fmaf(tv.x, tv.x, att); apt = fmaf(pv.x, tv.x, apt);
        app = fmaf(pv.y, pv.y, app); att = fmaf(tv.y, tv.y, att); apt = fmaf(pv.y, tv.y, apt);
        app = fmaf(pv.z, pv.z, app); att = fmaf(tv.z, tv.z, att); apt = fmaf(pv.z, tv.z, apt);
        app = fmaf(pv.w, pv.w, app); att = fmaf(tv.w, tv.w, att); apt = fmaf(pv.w, tv.w, apt);
      } else {
        float pa[4] = {pv.x, pv.y, pv.z, pv.w};
        float ta[4] = {tv.x, tv.y, tv.z, tv.w};
        int   ma[4] = {mv.x, mv.y, mv.z, mv.w};
#pragma unroll
        for (int j = 0; j < 4; ++j) {
          if (ma[j] != cur) {
            atomicAdd(&seg_pp[cur], app);
            atomicAdd(&seg_tt[cur], att);
            atomicAdd(&seg_pt[cur], apt);
            cur = ma[j];
            app = att = apt = 0.0f;
          }
          app = fmaf(pa[j], pa[j], app);
          att = fmaf(ta[j], ta[j], att);
          apt = fmaf(pa[j], ta[j], apt);
        }
      }
    }
  } else {
    for (int i = 0; i < count; ++i) {
      int   s  = bm[base + i];
      float pv = p[base + i];
      float tv = t[base + i];
      if (s != cur) {
        atomicAdd(&seg_pp[cur], app);
        atomicAdd(&seg_tt[cur], att);
        atomicAdd(&seg_pt[cur], apt);
        cur = s;
        app = att = apt = 0.0f;
      }
      app = fmaf(pv, pv, app);
      att = fmaf(tv, tv, att);
      apt = fmaf(pv, tv, apt);
    }
  }

  atomicAdd(&seg_pp[cur], app);
  atomicAdd(&seg_tt[cur], att);
  atomicAdd(&seg_pt[cur], apt);
}

// ---------------------------------------------------------------------------
// Stage 2: one wave32 block. Each lane accumulates two f32 partials of
// (1 - cosine); a single V_WMMA_F32_16X16X4_F32 against a ones B-matrix
// reduces all 64 lane-partials exactly in f32. Lane 0's c[0..7] = row sums
// 0..7, lane 16's c[0..7] = row sums 8..15 (CDNA5 16x16 f32 C/D layout).
// ---------------------------------------------------------------------------
__global__ __launch_bounds__(32) void cosine_mean_kernel(
    const float* __restrict__ seg_pp,
    const float* __restrict__ seg_tt,
    const float* __restrict__ seg_pt,
    float* __restrict__ out,
    int nseg) {
  const int lane = threadIdx.x;

  float acc0 = 0.0f, acc1 = 0.0f;
  // nseg % 64 == 0 => every lane runs identical trip counts (EXEC all-1s).
  for (int s = lane; s < nseg; s += 64) {
    float denom = fmaxf(sqrtf(seg_pp[s]) * sqrtf(seg_tt[s]), 1e-12f);
    acc0 += 1.0f - seg_pt[s] / denom;
  }
  for (int s = lane + 32; s < nseg; s += 64) {
    float denom = fmaxf(sqrtf(seg_pp[s]) * sqrtf(seg_tt[s]), 1e-12f);
    acc1 += 1.0f - seg_pt[s] / denom;
  }

  // A (16x4 f32): lane L in [0,16): VGPR0 = A[L,0], VGPR1 = A[L,1];
  //               lane L+16:        VGPR0 = A[L,2], VGPR1 = A[L,3].
  v2f a;
  a[0] = acc0;
  a[1] = acc1;
  v2f b;
  b[0] = 1.0f;   // ones 4x16 B-matrix => D[m,n] = sum_k A[m,k]
  b[1] = 1.0f;
  v8f c = {};
  c = __builtin_amdgcn_wmma_f32_16x16x4_f32(
      /*neg_a=*/false, a, /*neg_b=*/false, b,
      /*c_mod=*/(short)0, c, /*reuse_a=*/false, /*reuse_b=*/false);

  float s8 = c[0] + c[1] + c[2] + c[3] + c[4] + c[5] + c[6] + c[7];
  float total = __shfl(s8, 0, 32) + __shfl(s8, 16, 32);

  if (lane == 0) out[0] = total / (float)nseg;
}

extern "C" void kernel_launch(void* const* d_in, const int* in_sizes, int n_in,
                              void* d_out, int out_size, void* d_ws, size_t ws_size,
                              hipStream_t stream) {
  const float* preds  = (const float*)d_in[0];
  const float* target = (const float*)d_in[1];
  const int*   bm     = (const int*)d_in[2];  // integer input -> const int*
  float*       out    = (float*)d_out;

  const int n = in_sizes[0];

  float* seg_pp = (float*)d_ws;
  float* seg_tt = seg_pp + NSEG;
  float* seg_pt = seg_tt + NSEG;

  // Zero segment accumulators every call (capture-safe async memset).
  hipMemsetAsync(d_ws, 0, (size_t)3 * NSEG * sizeof(float), stream);

  long threads_needed = ((long)n + EPT - 1) / EPT;
  int  blocks = (int)((threads_needed + TPB - 1) / TPB);

  seg_reduce_kernel<<<blocks, TPB, 0, stream>>>(preds, target, bm,
                                                seg_pp, seg_tt, seg_pt, n);
  cosine_mean_kernel<<<1, 32, 0, stream>>>(seg_pp, seg_tt, seg_pt, out, NSEG);
}